// RuleBasedGatingNetwork_44057774523075
// MI455X (gfx1250) — compile-verified
//
#include <hip/hip_runtime.h>
#include <stdint.h>

// RuleBasedGatingNetwork for MI455X (gfx1250).
// Pure HBM-streaming problem: ~304 MB moved, ~40 MFLOP -> bandwidth bound
// (~13 us floor at 23.3 TB/s). Strategy: gfx1250 async global->LDS copies
// (ASYNCcnt-tracked, double buffered, GVS saddr addressing to keep address
// math scalar), 128-bit non-temporal stores for the write-once output.

typedef __attribute__((ext_vector_type(4))) float v4f;

#define BLOCK        256
#define RPT          4                       // rows per thread per tile
#define TILE_ROWS    (BLOCK * RPT)           // 1024 rows / tile
#define TILES_PER_WG 4
#define WG_ROWS      (TILE_ROWS * TILES_PER_WG)  // 4096 rows / workgroup

__device__ __forceinline__ void compute_gate(float c4, float c5, float c6, float c7,
                                             float tr, float* w) {
    const float cyc = c4 + c5;
    const float hf  = c6 + c7;
    const bool  s0  = tr > 0.5f;          // expert 0
    const bool  sc  = cyc > hf;           // expert 1 vs 2
    w[0] = s0 ? 1.0f : 0.0f;
    w[1] = (!s0 && sc) ? 1.0f : 0.0f;
    w[2] = (!s0 && !sc) ? 1.0f : 0.0f;
}

__global__ __launch_bounds__(BLOCK)
void gating_async_kernel(const float* __restrict__ feat, float* __restrict__ out) {
    // Double-buffered staging: cols 4..7 packed as v4f, col 11 separately.
    // 2 * (16 KB + 4 KB) = 40 KB LDS -> several WGs per 320 KB WGP.
    __shared__ v4f   s_chf[2][TILE_ROWS];
    __shared__ float s_tr [2][TILE_ROWS];

    const int  t      = threadIdx.x;
    const long wgRow0 = (long)blockIdx.x * WG_ROWS;

    // 32-bit global byte offset of this thread's first row (row stride 64 B).
    // Max = 4M rows * 64 B = 2^28 -> fits the GVS signed-i32 VGPR offset.
    const unsigned gbase = (unsigned)((wgRow0 + (long)t * RPT) << 6);
    // Truncated generic address of a __shared__ object == LDS byte offset.
    const unsigned lchf0 = (unsigned)(uintptr_t)&s_chf[0][t * RPT];
    const unsigned ltr0  = (unsigned)(uintptr_t)&s_tr [0][t * RPT];

    // Issue one tile's async global->LDS copies: 8 ops per lane
    // (4x b128 for cols 4..7, 4x b32 for col 11), tracked by ASYNCcnt.
    // GVS form: uniform base pointer stays in SGPRs; per-op VGPR holds a
    // 32-bit byte offset (column offsets folded in, no imm offset -- the ISA
    // pseudocode adds INST_OFFSET to the LDS address too, so avoid it).
    auto issue_tile = [&](int tile) {
        const int      buf = tile & 1;
        const unsigned gt  = gbase + (unsigned)tile * (TILE_ROWS * 64);
#pragma unroll
        for (int k = 0; k < RPT; ++k) {
            const unsigned goff = gt + (unsigned)k * 64;
            const unsigned lchf = lchf0 + (unsigned)buf * (TILE_ROWS * 16) + (unsigned)k * 16;
            const unsigned ltr  = ltr0  + (unsigned)buf * (TILE_ROWS * 4)  + (unsigned)k * 4;
            asm volatile("global_load_async_to_lds_b128 %0, %1, %2"
                         :: "v"(lchf), "v"(goff + 16u), "s"(feat) : "memory");
            asm volatile("global_load_async_to_lds_b32 %0, %1, %2"
                         :: "v"(ltr), "v"(goff + 44u), "s"(feat) : "memory");
        }
    };

    // Consume one tile from LDS and emit 12 contiguous floats per thread as
    // three coalesced 16B-aligned non-temporal b128 stores (write-once output,
    // keep it out of the 192 MB L2 so the input stream owns the cache).
    auto process_tile = [&](int tile) {
        const int buf = tile & 1;
        float w[RPT][3];
#pragma unroll
        for (int k = 0; k < RPT; ++k) {
            const int   slot = t * RPT + k;
            const v4f   chf  = s_chf[buf][slot];
            const float tr   = s_tr [buf][slot];
            compute_gate(chf.x, chf.y, chf.z, chf.w, tr, w[k]);
        }
        v4f o0 = {w[0][0], w[0][1], w[0][2], w[1][0]};
        v4f o1 = {w[1][1], w[1][2], w[2][0], w[2][1]};
        v4f o2 = {w[2][2], w[3][0], w[3][1], w[3][2]};
        v4f* op = (v4f*)(out + (wgRow0 + (long)tile * TILE_ROWS + (long)t * RPT) * 3);
        __builtin_nontemporal_store(o0, op + 0);
        __builtin_nontemporal_store(o1, op + 1);
        __builtin_nontemporal_store(o2, op + 2);
    };

    // Software pipeline: two tiles (16 async ops) in flight before first use;
    // steady state = issue(t+1), wait "next tile's 8 still outstanding"
    // (ASYNCcnt retires in order), process(t).
    issue_tile(0);
    issue_tile(1);
    asm volatile("s_wait_asynccnt 0x8" ::: "memory");
    process_tile(0);
    issue_tile(2);
    asm volatile("s_wait_asynccnt 0x8" ::: "memory");
    process_tile(1);
    issue_tile(3);
    asm volatile("s_wait_asynccnt 0x8" ::: "memory");
    process_tile(2);
    asm volatile("s_wait_asynccnt 0x0" ::: "memory");
    process_tile(3);
}

// Scalar fallback for any rows not covered by the tiled kernel (none for the
// reference shape 4194304 = 1024 * WG_ROWS, but keeps the launch general).
__global__ void gating_tail_kernel(const float* __restrict__ feat,
                                   float* __restrict__ out,
                                   long start, long nRows) {
    long r = start + (long)blockIdx.x * blockDim.x + threadIdx.x;
    if (r >= nRows) return;
    const float* gp = feat + r * 16;
    float w[3];
    compute_gate(gp[4], gp[5], gp[6], gp[7], gp[11], w);
    out[r * 3 + 0] = w[0];
    out[r * 3 + 1] = w[1];
    out[r * 3 + 2] = w[2];
}

extern "C" void kernel_launch(void* const* d_in, const int* in_sizes, int n_in,
                              void* d_out, int out_size, void* d_ws, size_t ws_size,
                              hipStream_t stream) {
    const float* feat = (const float*)d_in[0];
    float*       out  = (float*)d_out;

    const long nRows    = (long)in_sizes[0] / 16;
    const long mainRows = (nRows / WG_ROWS) * WG_ROWS;

    if (mainRows > 0) {
        dim3 grid((unsigned)(mainRows / WG_ROWS));
        gating_async_kernel<<<grid, BLOCK, 0, stream>>>(feat, out);
    }
    const long rem = nRows - mainRows;
    if (rem > 0) {
        dim3 grid((unsigned)((rem + BLOCK - 1) / BLOCK));
        gating_tail_kernel<<<grid, BLOCK, 0, stream>>>(feat, out, mainRows, nRows);
    }
}